// segmentation_loss3D_22548578304824
// MI455X (gfx1250) — compile-verified
//
#include <hip/hip_runtime.h>
#include <math.h>

typedef float v2f __attribute__((ext_vector_type(2)));
typedef float v8f __attribute__((ext_vector_type(8)));

typedef __attribute__((address_space(1))) int* as1_int_ptr;
typedef __attribute__((address_space(3))) int* as3_int_ptr;

#define BIGF 1e9f
#define NELEM 4194304L    /* B*C*X*Y*Z = 2*8*64^3 */
#define NVOX  524288L     /* B*X*Y*Z */
#define NTOT  16777216L   /* 4 * NELEM (fg/bg for pred/target) */

// ---------------------------------------------------------------------------
// Zero the accumulator block (64 floats) + has_fg flags (4 ints).
// ---------------------------------------------------------------------------
__global__ void seg3d_zero_accum(float* __restrict__ accum, int* __restrict__ flags) {
    int t = threadIdx.x;
    if (t < 64) accum[t] = 0.0f;
    if (t < 4)  flags[t] = 0;
}

// ---------------------------------------------------------------------------
// Build the 4 mask f-fields (pred_fg, targ_fg, pred_bg, targ_bg) and has_fg.
// f = mask ? BIG : 0.   idx over (B,C,X,Y,Z); one thread per element.
// ---------------------------------------------------------------------------
__global__ void seg3d_init_masks(const float* __restrict__ x,
                                 const int*   __restrict__ tgt,
                                 float* __restrict__ f,
                                 int*   __restrict__ flags) {
    long idx = (long)blockIdx.x * 256 + threadIdx.x;          // < NELEM
    int  b   = (int)(idx >> 21);                              // / (8*64^3)
    int  c   = (int)((idx >> 18) & 7);                        // channel
    long vox = idx & 262143;                                  // within 64^3
    float xi = x[idx];
    int   t  = tgt[(((long)b) << 18) + vox];
    bool pfg = xi > 0.5f;                                     // pred foreground
    bool tfg = (t == c);                                      // one-hot > 0.5
    f[idx]             = pfg ? BIGF : 0.0f;
    f[NELEM + idx]     = tfg ? BIGF : 0.0f;
    f[2 * NELEM + idx] = pfg ? 0.0f : BIGF;
    f[3 * NELEM + idx] = tfg ? 0.0f : BIGF;
    // has_fg per (img, batch); a wave never straddles a batch boundary (2^21 % 32 == 0)
    unsigned long long mp = __ballot((int)pfg);
    unsigned long long mt = __ballot((int)tfg);
    if ((threadIdx.x & 31) == 0) {
        if (mp) atomicOr(&flags[b], 1);
        if (mt) atomicOr(&flags[2 + b], 1);
    }
}

// ---------------------------------------------------------------------------
// One exact 1D squared-EDT pass: d[i] = min_j f[j] + (i-j)^2 along an axis of
// the flat (4,B,C,X,Y,Z) array.  One block of 256 threads = (256/n) lines.
// Lines are staged into LDS via the gfx1250 async global->LDS path (ASYNCcnt)
// when available, then each output element runs an O(n) min-plus scan with
// broadcast LDS reads (conflict-free).
// ---------------------------------------------------------------------------
__global__ void seg3d_dt_pass(const float* __restrict__ in,
                              float* __restrict__ out,
                              int n, long stride) {
    __shared__ float lds[256];
    const int  t    = threadIdx.x;
    const int  lpb  = 256 / n;
    const int  l    = t / n;
    const int  i    = t - l * n;
    const long line = (long)blockIdx.x * lpb + l;
    const long o    = line / stride;
    const long r    = line - o * stride;
    const long base = o * (long)n * stride + r;
    const float* src = in + base + (long)i * stride;

#if __has_builtin(__builtin_amdgcn_global_load_async_to_lds_b32)
    {
        float* srcnc = const_cast<float*>(src);
        as1_int_ptr gp = (as1_int_ptr)srcnc;
        as3_int_ptr lp = (as3_int_ptr)&lds[t];
        __builtin_amdgcn_global_load_async_to_lds_b32(gp, lp, /*offset=*/0, /*cpol=*/0);
    }
#else
    {
        unsigned long long ga = (unsigned long long)src;
        unsigned la = (unsigned)(unsigned long long)
                      (__attribute__((address_space(3))) float*)&lds[t];
        asm volatile("global_load_async_to_lds_b32 %0, %1, off"
                     :: "v"(la), "v"(ga) : "memory");
    }
#endif
#if __has_builtin(__builtin_amdgcn_s_wait_asynccnt)
    __builtin_amdgcn_s_wait_asynccnt(0);
#else
    asm volatile("s_wait_asynccnt 0x0" ::: "memory");
#endif
    __syncthreads();

    const float* lf = lds + l * n;
    float m = 3e30f;
    for (int j = 0; j < n; ++j) {
        float dj = (float)(i - j);
        m = fminf(m, lf[j] + dj * dj);
    }
    out[base + (long)i * stride] = m;
}

// ---------------------------------------------------------------------------
// Cross-entropy + Tversky partials.  One thread per voxel of (B,X,Y,Z).
// accum layout (floats): [0..15]=sum(probs)[b,c] [16..31]=tps [32..47]=count
// [48]=sum(logp@target) [49]=hd_sum
// ---------------------------------------------------------------------------
__global__ void seg3d_ce_tversky(const float* __restrict__ x,
                                 const int*   __restrict__ tgt,
                                 float* __restrict__ accum) {
    __shared__ float sacc[49];
    const int tid = threadIdx.x;
    if (tid < 49) sacc[tid] = 0.0f;
    __syncthreads();

    long v   = (long)blockIdx.x * 256 + tid;      // < NVOX
    int  b   = (int)(v >> 18);
    long vox = v & 262143;
    const float* xp = x + (((long)b) << 21) + vox;

    float lg[8];
    float mx = -3e30f;
#pragma unroll
    for (int c = 0; c < 8; ++c) {
        lg[c] = xp[((long)c) << 18];
        mx = fmaxf(mx, lg[c]);
    }
    float s = 0.0f;
#pragma unroll
    for (int c = 0; c < 8; ++c) s += __expf(lg[c] - mx);
    float inv  = 1.0f / s;
    float logZ = mx + __logf(s);
    int   t    = tgt[v];

#pragma unroll
    for (int c = 0; c < 8; ++c)
        atomicAdd(&sacc[b * 8 + c], __expf(lg[c] - mx) * inv);   // sum probs
    atomicAdd(&sacc[16 + b * 8 + t], __expf(lg[t] - mx) * inv);  // tps
    atomicAdd(&sacc[32 + b * 8 + t], 1.0f);                      // count
    atomicAdd(&sacc[48], lg[t] - logZ);                          // sum logp

    __syncthreads();
    if (tid < 49) atomicAdd(&accum[tid], sacc[tid]);
}

// ---------------------------------------------------------------------------
// Hausdorff term: sum( (x - onehot)^2 * (pred_dt^2 + targ_dt^2) ).
// Each wave packs 64 products per step into an A tile and accumulates with
// V_WMMA_F32_16X16X4_F32 against an all-ones B (matrix-pipe sum engine).
// Grid 512x256 => 131072 lanes, uniform 16-iteration loop, EXEC all-1s.
// ---------------------------------------------------------------------------
__global__ void seg3d_hd_wmma(const float* __restrict__ x,
                              const int*   __restrict__ tgt,
                              const float* __restrict__ f,
                              const int*   __restrict__ flags,
                              float* __restrict__ hd_sum) {
    const long lane    = (long)blockIdx.x * 256 + threadIdx.x;  // 0..131071
    const long stride2 = 131072L * 2;                           // elems/iteration

    v8f acc = {};
    v2f ones;
    ones[0] = 1.0f;
    ones[1] = 1.0f;

    const float pf0 = flags[0] ? 1.0f : 0.0f;
    const float pf1 = flags[1] ? 1.0f : 0.0f;
    const float tf0 = flags[2] ? 1.0f : 0.0f;
    const float tf1 = flags[3] ? 1.0f : 0.0f;

    for (int it = 0; it < 16; ++it) {
        v2f a;
#pragma unroll
        for (int u = 0; u < 2; ++u) {
            long idx = (long)it * stride2 + lane * 2 + u;       // < NELEM
            int  b   = (int)(idx >> 21);
            int  c   = (int)((idx >> 18) & 7);
            long vox = idx & 262143;
            int  t   = tgt[(((long)b) << 18) + vox];
            float oh = (t == c) ? 1.0f : 0.0f;
            float xi = x[idx];
            float e  = (xi - oh) * (xi - oh);
            float pd = sqrtf(f[idx]) + sqrtf(f[2 * NELEM + idx]);
            float td = sqrtf(f[NELEM + idx]) + sqrtf(f[3 * NELEM + idx]);
            pd *= (b ? pf1 : pf0);                              // has_fg guard
            td *= (b ? tf1 : tf0);
            a[u] = e * (pd * pd + td * td);
            // stream-ahead hint for next iteration's field reads
            __builtin_prefetch(&f[idx + stride2], 0, 1);
        }
        // D[m][n] = sum_k A[m][k]*1 + C[m][n]  -> 64-wide sum per instruction
        acc = __builtin_amdgcn_wmma_f32_16x16x4_f32(
            /*neg_a=*/false, a, /*neg_b=*/false, ones,
            /*c_mod=*/(short)0, acc, /*reuse_a=*/false, /*reuse_b=*/false);
    }

    float s = 0.0f;
#pragma unroll
    for (int k = 0; k < 8; ++k) s += acc[k];
    for (int off = 16; off > 0; off >>= 1) s += __shfl_xor(s, off, 32);
    // every column of D holds the same row sums -> divide by 16
    if ((threadIdx.x & 31) == 0) atomicAdd(hd_sum, s * (1.0f / 16.0f));
}

// ---------------------------------------------------------------------------
// Combine partials into the scalar loss.
// ---------------------------------------------------------------------------
__global__ void seg3d_finalize(const float* __restrict__ accum,
                               float* __restrict__ out) {
    float mean_score = 0.0f;
    for (int c = 1; c < 8; ++c) {
        float sc = 0.0f;
        for (int b = 0; b < 2; ++b) {
            float sp  = accum[b * 8 + c];
            float tp  = accum[16 + b * 8 + c];
            float cnt = accum[32 + b * 8 + c];
            float fps = (sp - tp) * 1.0f;   // ALPHA_T = 1.0
            float fns = (cnt - tp) * 1.0f;  // BETA_T  = 1.0
            sc += 2.0f * tp / (2.0f * tp + fps + fns + 1e-5f);
        }
        mean_score += sc * 0.5f;            // mean over batch
    }
    float dice = 1.0f - mean_score / 7.0f;  // mean over channels 1..7
    float ce   = -accum[48] / (float)NVOX;
    float hd   =  accum[49] / (float)NELEM;
    out[0] = dice + ce + hd;
}

// ---------------------------------------------------------------------------
extern "C" void kernel_launch(void* const* d_in, const int* in_sizes, int n_in,
                              void* d_out, int out_size, void* d_ws, size_t ws_size,
                              hipStream_t stream) {
    const float* x   = (const float*)d_in[0];   // (2,8,64,64,64) f32 logits
    const int*   tgt = (const int*)d_in[1];     // (2,64,64,64)   i32 labels
    float*       out = (float*)d_out;

    float* bufA  = (float*)d_ws;                // 4*(B,C,X,Y,Z) f-fields
    float* bufB  = bufA + NTOT;                 // ping-pong buffer
    float* accum = bufB + NTOT;                 // 64 floats of partials
    int*   flags = (int*)(accum + 64);          // has_fg[2][2]

    seg3d_zero_accum<<<1, 128, 0, stream>>>(accum, flags);
    seg3d_init_masks<<<NELEM / 256, 256, 0, stream>>>(x, tgt, bufA, flags);

    // Exact EDT over axes (C, X, Y, Z) of the (4,B,C,X,Y,Z) stack, matching
    // the reference's min-plus scan order.  All pass grids are exactly 65536.
    seg3d_dt_pass<<<65536, 256, 0, stream>>>(bufA, bufB, 8,  262144L); // C
    seg3d_dt_pass<<<65536, 256, 0, stream>>>(bufB, bufA, 64, 4096L);   // X
    seg3d_dt_pass<<<65536, 256, 0, stream>>>(bufA, bufB, 64, 64L);     // Y
    seg3d_dt_pass<<<65536, 256, 0, stream>>>(bufB, bufA, 64, 1L);      // Z

    seg3d_ce_tversky<<<NVOX / 256, 256, 0, stream>>>(x, tgt, accum);
    seg3d_hd_wmma<<<512, 256, 0, stream>>>(x, tgt, bufA, flags, accum + 49);
    seg3d_finalize<<<1, 1, 0, stream>>>(accum, out);
}